// SolarSpringAttention_88158498717859
// MI455X (gfx1250) — compile-verified
//
#include <hip/hip_runtime.h>
#include <math.h>

#define LTOK 4096
#define DDIM 2048

#define BM 128
#define BN 64
#define BK 32

typedef float v2f __attribute__((ext_vector_type(2)));
typedef float v8f __attribute__((ext_vector_type(8)));
typedef unsigned int v4u __attribute__((ext_vector_type(4)));
typedef int v8i_t __attribute__((ext_vector_type(8)));
typedef int v4i_t __attribute__((ext_vector_type(4)));

// ---------------------------------------------------------------------------
// Kernel 1: semantic mass m[i] = ||v_i|| * POS_W[p] * (1 + res), black-hole bh[i]
// ---------------------------------------------------------------------------
__global__ __launch_bounds__(256) void mass_bh_kernel(
    const float* __restrict__ tv, const int* __restrict__ pos,
    const float* __restrict__ conf, const float* __restrict__ res,
    float* __restrict__ m, float* __restrict__ bh) {
  __shared__ float rbuf[256];
  const int i = blockIdx.x;
  const int tid = threadIdx.x;
  const float* row = tv + (size_t)i * DDIM;
  float s = 0.f;
  for (int j = tid; j < DDIM; j += 256) {
    float v = row[j];
    s += v * v;
  }
  rbuf[tid] = s;
  __syncthreads();
  for (int k = 128; k > 0; k >>= 1) {
    if (tid < k) rbuf[tid] += rbuf[tid + k];
    __syncthreads();
  }
  if (tid == 0) {
    const float POS_W[8] = {0.95f, 0.9f, 0.85f, 0.5f, 0.4f, 0.2f, 0.15f, 0.05f};
    float norm = sqrtf(rbuf[0]);
    int p = pos[i] & 7;
    m[i] = norm * POS_W[p] * (1.f + res[i]);
    float c = conf[i];
    if (c <= 0.1f) {
      bh[i] = -1.0e6f;
    } else {
      float g = c - 0.1f;
      bh[i] = -0.01f / (g * g);
    }
  }
}

// ---------------------------------------------------------------------------
// Kernel 2: scores row + softmax row (block per row i). 16KB LDS row buffer.
// ---------------------------------------------------------------------------
__global__ __launch_bounds__(256) void scores_softmax_kernel(
    const float* __restrict__ m, const float* __restrict__ bh,
    const int* __restrict__ pos, const int* __restrict__ depth,
    const int* __restrict__ tpos, const int* __restrict__ slot,
    const float* __restrict__ Gmic, const float* __restrict__ Gmac,
    const float* __restrict__ kspr, const float* __restrict__ temp,
    float* __restrict__ scores, float* __restrict__ Aout) {
  __shared__ float sc[LTOK];
  __shared__ float sgm[8];
  __shared__ float rbuf[256];

  const int i = blockIdx.x;
  const int tid = threadIdx.x;
  const int p_i = pos[i] & 7;

  if (tid < 8) sgm[tid] = 1.f / (1.f + expf(-Gmic[p_i * 8 + tid]));
  __syncthreads();

  const float m_i = m[i];
  const float bh_i = bh[i];
  const int slot_i = slot[i];
  const int depth_i = depth[i];
  const int tp_i = tpos[i];
  const float ksig = 1.f / (1.f + expf(-kspr[p_i]));
  const float gmac = 1.f / (1.f + expf(-Gmac[0]));
  const bool rowskip = (bh_i < -1.0e5f);

  for (int j = tid; j < LTOK; j += 256) {
    float s = 0.f;
    if (!rowskip && j != i) {
      int p_j = pos[j] & 7;
      float mm = m_i * m[j];
      float rs = fmaxf(fabsf((float)(slot_i - slot[j])), 1.f);
      float f_micro = sgm[p_j] * mm / (rs * rs);
      float ro = fabsf((float)(depth_i - depth[j])) + 1.f;
      float f_macro = gmac * mm / (ro * ro);
      float f_spring = ksig * fabsf((float)(tp_i - tpos[j]));
      s = f_micro + f_macro + f_spring + bh_i + bh[j];
    }
    sc[j] = s;
    scores[(size_t)i * LTOK + j] = s;
  }
  __syncthreads();

  float mx = -3.4e38f;
  for (int j = tid; j < LTOK; j += 256) mx = fmaxf(mx, sc[j]);
  rbuf[tid] = mx;
  __syncthreads();
  for (int k = 128; k > 0; k >>= 1) {
    if (tid < k) rbuf[tid] = fmaxf(rbuf[tid], rbuf[tid + k]);
    __syncthreads();
  }
  mx = rbuf[0];
  __syncthreads();

  const float T = fmaxf(fabsf(temp[0]), 0.1f);
  const float invT = 1.f / T;
  float lsum = 0.f;
  for (int j = tid; j < LTOK; j += 256) {
    float e = expf((sc[j] - mx) * invT);
    sc[j] = e;
    lsum += e;
  }
  rbuf[tid] = lsum;
  __syncthreads();
  for (int k = 128; k > 0; k >>= 1) {
    if (tid < k) rbuf[tid] += rbuf[tid + k];
    __syncthreads();
  }
  const float inv = 1.f / rbuf[0];
  for (int j = tid; j < LTOK; j += 256) {
    Aout[(size_t)i * LTOK + j] = sc[j] * inv;
  }
}

// ---------------------------------------------------------------------------
// TDM helper: issue a 2D tensor_load_to_lds.
//   tile_d0 = contiguous elements per row, tile_d1 = rows,
//   row_stride = global row stride (elements), fp32 data (data_size=2 -> 4B).
// D# packing per CDNA5 ISA ch.8 (group0 128b, group1 256b, groups2/3 unused).
// 6-arg builtin variant (clang-23 / therock-10.0 headers).
// ---------------------------------------------------------------------------
__device__ __forceinline__ void tdm_load_2d(unsigned lds_off, const float* gptr,
                                            unsigned tile_d0, unsigned tile_d1,
                                            unsigned long long row_stride) {
  unsigned long long ga = (unsigned long long)(size_t)gptr;

  v4u g0;
  g0[0] = 1u;                                       // count=1, user descriptor
  g0[1] = lds_off;                                  // lds_addr (bytes)
  g0[2] = (unsigned)(ga & 0xffffffffull);           // global_addr[31:0]
  g0[3] = (unsigned)((ga >> 32) & 0x1ffffffull)     // global_addr[56:32]
          | (2u << 30);                             // type=2 ("image")

  const unsigned td0 = tile_d0;                     // tensor_dim0 (no clipping)
  const unsigned td1 = tile_d1;                     // tensor_dim1 (no clipping)
  v8i_t g1;
  g1[0] = (int)(2u << 16);                          // wg_mask=0, data_size=2 (4B)
  g1[1] = (int)((td0 & 0xffffu) << 16);             // tensor_dim0[15:0]
  g1[2] = (int)(((td0 >> 16) & 0xffffu) | ((td1 & 0xffffu) << 16));
  g1[3] = (int)(((td1 >> 16) & 0xffffu) | ((tile_d0 & 0xffffu) << 16)); // tile_dim0
  g1[4] = (int)(tile_d1 & 0xffffu);                 // tile_dim1, tile_dim2=0
  g1[5] = (int)(unsigned)(row_stride & 0xffffffffull);        // dim0_stride[31:0]
  g1[6] = (int)(unsigned)((row_stride >> 32) & 0xffffull);    // dim0_stride[47:32]
  g1[7] = 0;

  v4i_t z4 = {0, 0, 0, 0};
  v8i_t z8 = {0, 0, 0, 0, 0, 0, 0, 0};
  __builtin_amdgcn_tensor_load_to_lds(g0, g1, z4, z4, z8, 0);
}

// ---------------------------------------------------------------------------
// Kernel 3: f32 WMMA GEMM, TDM double-buffered through LDS.
//   C[MxN] = A[MxK] * B (+bias)
//   TRANSB=1: B is NxK row-major (B^T applied)  -> LDS B tile [BN][BK]
//   TRANSB=0: B is KxN row-major                -> LDS B tile [BK][BN]
// Block: 256 threads = 8 waves; block tile BM x BN = 128 x 64;
// wave w computes rows [w*16, w*16+16) x all 64 cols (4 accumulators).
// A frag (16x4 f32): lane = M%16 + 16*(K-pair), v2f = {K, K+1}.
// C/D frag: VGPR r holds row M = r + 8*(lane/16), col N = lane%16.
// ---------------------------------------------------------------------------
template <bool TRANSB, bool BIAS>
__global__ __launch_bounds__(256) void wmma_gemm_f32(
    const float* __restrict__ A, const float* __restrict__ B,
    const float* __restrict__ bias, float* __restrict__ C,
    int M, int N, int K) {
  __shared__ float sA[2][BM * BK];   // [row][k], 16KB per buffer
  __shared__ float sB[2][BN * BK];   // TRANSB: [col][k]; else [k][col], 8KB each

  const int tid = threadIdx.x;
  const int wave = tid >> 5;
  const int lane = tid & 31;
  const int half = lane >> 4;        // selects K pair {0,1} vs {2,3}
  const int l16 = lane & 15;
  const int row0 = blockIdx.y * BM;
  const int col0 = blockIdx.x * BN;

  v8f acc0 = {}, acc1 = {}, acc2 = {}, acc3 = {};

  const int nchunk = K / BK;

  // Prologue: wave 0 DMAs chunk 0 into buffer 0.
  if (tid < 32) {
    tdm_load_2d((unsigned)(size_t)&sA[0][0], A + (size_t)row0 * K,
                BK, BM, (unsigned long long)K);
    if (TRANSB)
      tdm_load_2d((unsigned)(size_t)&sB[0][0], B + (size_t)col0 * K,
                  BK, BN, (unsigned long long)K);
    else
      tdm_load_2d((unsigned)(size_t)&sB[0][0], B + col0,
                  BN, BK, (unsigned long long)N);
  }

  for (int c = 0; c < nchunk; ++c) {
    const int cur = c & 1;
    if (tid < 32) __builtin_amdgcn_s_wait_tensorcnt(0);
    __syncthreads();   // chunk c resident; all waves done with buffer cur^1

    if ((tid < 32) && (c + 1 < nchunk)) {
      const int nb = cur ^ 1;
      const int koff = (c + 1) * BK;
      tdm_load_2d((unsigned)(size_t)&sA[nb][0], A + (size_t)row0 * K + koff,
                  BK, BM, (unsigned long long)K);
      if (TRANSB)
        tdm_load_2d((unsigned)(size_t)&sB[nb][0], B + (size_t)col0 * K + koff,
                    BK, BN, (unsigned long long)K);
      else
        tdm_load_2d((unsigned)(size_t)&sB[nb][0], B + (size_t)koff * N + col0,
                    BN, BK, (unsigned long long)N);
    }

    const float* __restrict__ a_t = &sA[cur][(wave * 16 + l16) * BK];
    const float* __restrict__ b_t = &sB[cur][0];

#pragma unroll
    for (int kk = 0; kk < BK; kk += 4) {
      const int ka = kk + 2 * half;
      v2f a;  a.x = a_t[ka]; a.y = a_t[ka + 1];
      v2f b0, b1, b2, b3;
      if (TRANSB) {
        const float* p0 = b_t + l16 * BK + ka;
        b0.x = p0[0];        b0.y = p0[1];
        b1.x = p0[16 * BK];  b1.y = p0[16 * BK + 1];
        b2.x = p0[32 * BK];  b2.y = p0[32 * BK + 1];
        b3.x = p0[48 * BK];  b3.y = p0[48 * BK + 1];
      } else {
        const float* r0 = b_t + ka * BN + l16;
        const float* r1 = b_t + (ka + 1) * BN + l16;
        b0.x = r0[0];  b0.y = r1[0];
        b1.x = r0[16]; b1.y = r1[16];
        b2.x = r0[32]; b2.y = r1[32];
        b3.x = r0[48]; b3.y = r1[48];
      }
      acc0 = __builtin_amdgcn_wmma_f32_16x16x4_f32(false, a, false, b0, (short)0, acc0, false, false);
      acc1 = __builtin_amdgcn_wmma_f32_16x16x4_f32(false, a, false, b1, (short)0, acc1, false, false);
      acc2 = __builtin_amdgcn_wmma_f32_16x16x4_f32(false, a, false, b2, (short)0, acc2, false, false);
      acc3 = __builtin_amdgcn_wmma_f32_16x16x4_f32(false, a, false, b3, (short)0, acc3, false, false);
    }
    __syncthreads();   // all reads of buffer cur done before it is refilled
  }

  float bv0 = 0.f, bv1 = 0.f, bv2 = 0.f, bv3 = 0.f;
  if (BIAS) {
    bv0 = bias[col0 + l16];
    bv1 = bias[col0 + 16 + l16];
    bv2 = bias[col0 + 32 + l16];
    bv3 = bias[col0 + 48 + l16];
  }

#pragma unroll
  for (int r = 0; r < 8; ++r) {
    const int rr = row0 + wave * 16 + 8 * half + r;
    float* Crow = C + (size_t)rr * N + col0 + l16;
    Crow[0]  = acc0[r] + bv0;
    Crow[16] = acc1[r] + bv1;
    Crow[32] = acc2[r] + bv2;
    Crow[48] = acc3[r] + bv3;
  }
}

// ---------------------------------------------------------------------------
// Host-side orchestration
// ---------------------------------------------------------------------------
extern "C" void kernel_launch(void* const* d_in, const int* in_sizes, int n_in,
                              void* d_out, int out_size, void* d_ws, size_t ws_size,
                              hipStream_t stream) {
  (void)in_sizes; (void)n_in; (void)out_size; (void)ws_size;

  const float* tv  = (const float*)d_in[0];   // [L,D]
  const int*   pos = (const int*)d_in[1];     // [L]
  const int*   dep = (const int*)d_in[2];     // [L]
  const int*   tps = (const int*)d_in[3];     // [L]
  const int*   slt = (const int*)d_in[4];     // [L]
  const float* cnf = (const float*)d_in[5];   // [L]
  const float* res = (const float*)d_in[6];   // [L]
  const float* Gmi = (const float*)d_in[7];   // [8,8]
  const float* Gma = (const float*)d_in[8];   // [1]
  const float* ksp = (const float*)d_in[9];   // [8]
  const float* tmp = (const float*)d_in[10];  // [1]
  const float* Wv  = (const float*)d_in[11];  // [D,D]
  const float* Wvb = (const float*)d_in[12];  // [D]
  const float* Wo  = (const float*)d_in[13];  // [D,D]
  const float* Wob = (const float*)d_in[14];  // [D]

  // Output tuple laid out flat: out [L*D], A [L*L], scores [L*L]
  float* outp = (float*)d_out;
  float* Amat = outp + (size_t)LTOK * DDIM;
  float* Smat = Amat + (size_t)LTOK * LTOK;

  // Workspace: m [L], bh [L], V [L*D], AV [L*D]
  float* ws   = (float*)d_ws;
  float* mArr = ws;
  float* bhA  = ws + LTOK;
  float* V    = ws + 2 * LTOK;
  float* AV   = V + (size_t)LTOK * DDIM;

  // 1) masses + black-hole terms
  mass_bh_kernel<<<LTOK, 256, 0, stream>>>(tv, pos, cnf, res, mArr, bhA);

  // 2) scores + softmax (writes Smat and Amat)
  scores_softmax_kernel<<<LTOK, 256, 0, stream>>>(mArr, bhA, pos, dep, tps, slt,
                                                  Gmi, Gma, ksp, tmp, Smat, Amat);

  dim3 g(DDIM / BN, LTOK / BM);   // (32, 32)

  // 3) V = tv @ Wv^T + Wv_b      (M=L, N=D, K=D, B transposed)
  wmma_gemm_f32<true, true><<<g, 256, 0, stream>>>(tv, Wv, Wvb, V, LTOK, DDIM, DDIM);

  // 4) AV = A @ V                (M=L, N=D, K=L, B row-major)
  wmma_gemm_f32<false, false><<<g, 256, 0, stream>>>(Amat, V, (const float*)nullptr,
                                                     AV, LTOK, DDIM, LTOK);

  // 5) out = AV @ Wo^T + Wo_b    (M=L, N=D, K=D, B transposed)
  wmma_gemm_f32<true, true><<<g, 256, 0, stream>>>(AV, Wo, Wob, outp, LTOK, DDIM, DDIM);
}